// DownSampleBlock_34213709480334
// MI455X (gfx1250) — compile-verified
//
#include <hip/hip_runtime.h>
#include <stdint.h>

#define THREADS 256
#define EDGE_TILE 1024
#define SORT_TILE 2048   // elements per radix block
#define SORT_RPT 8       // rounds per block (SORT_TILE / 256)

// ---------------------------------------------------------------------------
// K1: per-node sort key.  score = tanh((x.w)/||w||) is monotone in dot = x.w,
// so we sort on dot only.  Key transform -> ascending-uint sort == descending
// float order; stable LSD radix -> ties resolve by lower index first,
// matching jax.lax.top_k.
// ---------------------------------------------------------------------------
__global__ void score_kernel(const float* __restrict__ x, const float* __restrict__ w,
                             unsigned* __restrict__ keys, int* __restrict__ vals, int n) {
    int i = blockIdx.x * blockDim.x + threadIdx.x;
    if (i >= n) return;
    float w0 = w[0], w1 = w[1], w2 = w[2];
    float d = fmaf(x[3 * i], w0, fmaf(x[3 * i + 1], w1, x[3 * i + 2] * w2));
    unsigned u = __float_as_uint(d);
    unsigned ka = (u & 0x80000000u) ? ~u : (u | 0x80000000u);  // ascending-float key
    keys[i] = ~ka;                                             // descending order
    vals[i] = i;
}

// ---------------------------------------------------------------------------
// Radix pass, 2048-element tiles: digit-major histogram hist[d * nb + b],
// tiny single-block scan (nb ~ 489), stable multi-round ballot scatter.
// ---------------------------------------------------------------------------
__global__ void hist_kernel(const unsigned* __restrict__ keys, unsigned* __restrict__ hist,
                            int n, int nb, int shift) {
    __shared__ unsigned lh[256];
    int t = threadIdx.x;
    lh[t] = 0;
    __syncthreads();
    int base = blockIdx.x * SORT_TILE;
#pragma unroll
    for (int r = 0; r < SORT_RPT; ++r) {
        int i = base + r * 256 + t;
        if (i < n) atomicAdd(&lh[(keys[i] >> shift) & 0xFFu], 1u);
    }
    __syncthreads();
    hist[t * nb + blockIdx.x] = lh[t];
}

__global__ void scan_kernel(unsigned* __restrict__ hist, int nb) {
    __shared__ unsigned tot[256];
    __shared__ unsigned base[256];
    int d = threadIdx.x;
    unsigned s = 0;
    for (int b = 0; b < nb; ++b) s += hist[d * nb + b];
    tot[d] = s;
    __syncthreads();
    if (d == 0) {
        unsigned run = 0;
        for (int j = 0; j < 256; ++j) { base[j] = run; run += tot[j]; }
    }
    __syncthreads();
    unsigned run = base[d];
    for (int b = 0; b < nb; ++b) {
        unsigned c = hist[d * nb + b];
        hist[d * nb + b] = run;   // exclusive global start for (digit d, block b)
        run += c;
    }
}

// Stable scatter: 8 rounds x 256 lanes.  Within a round, wave32 ballot
// multisplit ranks elements; wcnt[wave][digit] orders across waves;
// running[digit] carries counts across rounds.  Order = round, wave, lane.
__global__ void scatter_kernel(const unsigned* __restrict__ kin, const int* __restrict__ vin,
                               unsigned* __restrict__ kout, int* __restrict__ vout,
                               const unsigned* __restrict__ hist, int n, int nb, int shift) {
    __shared__ unsigned wcnt[8][256];
    __shared__ unsigned running[256];
    int t = threadIdx.x;
    int wv = t >> 5;
    unsigned ln = (unsigned)(t & 31);
    running[t] = 0;
    int base = blockIdx.x * SORT_TILE;

    for (int r = 0; r < SORT_RPT; ++r) {
#pragma unroll
        for (int w = 0; w < 8; ++w) wcnt[w][t] = 0;
        __syncthreads();  // wcnt zeroed; prior running[] update visible

        int i = base + r * 256 + t;
        bool valid = (i < n);
        unsigned key = 0, dig = 0;
        int val = 0;
        if (valid) { key = kin[i]; val = vin[i]; dig = (key >> shift) & 0xFFu; }

        unsigned active = __builtin_amdgcn_ballot_w32(valid);
        unsigned peers = active;
#pragma unroll
        for (int b = 0; b < 8; ++b) {
            unsigned bb = __builtin_amdgcn_ballot_w32(((dig >> b) & 1u) != 0u);
            peers &= ((dig >> b) & 1u) ? bb : ~bb;
        }
        unsigned rank = 0;
        if (valid) {
            unsigned lt = (ln == 0) ? 0u : ((1u << ln) - 1u);
            rank = (unsigned)__popc(peers & lt);
            if (rank == 0) wcnt[wv][dig] = (unsigned)__popc(peers);  // group leader
        }
        __syncthreads();  // wcnt complete

        if (valid) {
            unsigned off = running[dig] + rank;
            for (int w = 0; w < wv; ++w) off += wcnt[w][dig];
            unsigned pos = hist[dig * nb + blockIdx.x] + off;
            kout[pos] = key;
            vout[pos] = val;
        }
        unsigned tot = 0;
#pragma unroll
        for (int w = 0; w < 8; ++w) tot += wcnt[w][t];
        __syncthreads();  // all running[] readers done
        running[t] += tot;
    }
}

// ---------------------------------------------------------------------------
// K3: gather top-k features, gate by tanh score, build relabel table.
// vals[] is a permutation of [0,n), so every new_idx entry is written once.
// ---------------------------------------------------------------------------
__global__ void gather_kernel(const float* __restrict__ x, const float* __restrict__ w,
                              const int* __restrict__ vals, float* __restrict__ xout,
                              int* __restrict__ new_idx, int n, int k) {
    int i = blockIdx.x * blockDim.x + threadIdx.x;
    if (i >= n) return;
    int idx = vals[i];
    if (i < k) {
        float w0 = w[0], w1 = w[1], w2 = w[2];
        float invn = rsqrtf(fmaf(w0, w0, fmaf(w1, w1, w2 * w2)));
        float a = x[3 * idx], b = x[3 * idx + 1], c = x[3 * idx + 2];
        float s = tanhf(fmaf(a, w0, fmaf(b, w1, c * w2)) * invn);
        xout[3 * i + 0] = a * s;
        xout[3 * i + 1] = b * s;
        xout[3 * i + 2] = c * s;
        new_idx[idx] = i;
    } else {
        new_idx[idx] = -1;
    }
}

// ---------------------------------------------------------------------------
// K4: edge relabel.  Streams the two edge-index rows into LDS with the CDNA5
// async global->LDS path (ASYNCcnt), then gathers from the 4MB relabel table
// (resident in the 192MB L2) and emits float4 stores.
// ---------------------------------------------------------------------------
__global__ void edge_kernel(const int* __restrict__ esrc, const int* __restrict__ edst,
                            const int* __restrict__ relab,
                            float* __restrict__ o0, float* __restrict__ o1,
                            float* __restrict__ om, int E) {
    __shared__ int smem[2 * EDGE_TILE];  // 4KB src tile + 4KB dst tile
    const int t = threadIdx.x;
    const int tile0 = blockIdx.x * EDGE_TILE;

    if (tile0 + EDGE_TILE <= E) {
        unsigned lds_base = (unsigned)(uintptr_t)(void*)smem;
        unsigned goff = (unsigned)(tile0 + t * 4) * 4u;  // byte offset, 16B per lane
        // gfx1250 async bulk copy: 32 lanes x 16B per instruction, no VGPR round-trip
        asm volatile("global_load_async_to_lds_b128 %0, %1, %2"
                     :: "v"(lds_base + (unsigned)t * 16u), "v"(goff), "s"(esrc)
                     : "memory");
        asm volatile("global_load_async_to_lds_b128 %0, %1, %2"
                     :: "v"(lds_base + (unsigned)(EDGE_TILE * 4) + (unsigned)t * 16u),
                        "v"(goff), "s"(edst)
                     : "memory");
        asm volatile("s_wait_asynccnt 0" ::: "memory");
        __syncthreads();

        int4 s4 = ((const int4*)smem)[t];
        int4 d4 = ((const int4*)smem)[EDGE_TILE / 4 + t];

        int a0 = relab[s4.x], a1 = relab[s4.y], a2 = relab[s4.z], a3 = relab[s4.w];
        int b0 = relab[d4.x], b1 = relab[d4.y], b2 = relab[d4.z], b3 = relab[d4.w];
        bool v0 = (a0 >= 0) && (b0 >= 0);
        bool v1 = (a1 >= 0) && (b1 >= 0);
        bool v2 = (a2 >= 0) && (b2 >= 0);
        bool v3 = (a3 >= 0) && (b3 >= 0);

        float4 r0 = make_float4(v0 ? (float)a0 : -1.f, v1 ? (float)a1 : -1.f,
                                v2 ? (float)a2 : -1.f, v3 ? (float)a3 : -1.f);
        float4 r1 = make_float4(v0 ? (float)b0 : -1.f, v1 ? (float)b1 : -1.f,
                                v2 ? (float)b2 : -1.f, v3 ? (float)b3 : -1.f);
        float4 rm = make_float4(v0 ? 1.f : 0.f, v1 ? 1.f : 0.f,
                                v2 ? 1.f : 0.f, v3 ? 1.f : 0.f);

        int e0 = tile0 + t * 4;
        if (((((uintptr_t)o0) | ((uintptr_t)o1) | ((uintptr_t)om)) & 15) == 0) {
            *(float4*)(o0 + e0) = r0;   // 16B global_store_b128
            *(float4*)(o1 + e0) = r1;
            *(float4*)(om + e0) = rm;
        } else {
            float r0a[4] = {r0.x, r0.y, r0.z, r0.w};
            float r1a[4] = {r1.x, r1.y, r1.z, r1.w};
            float rma[4] = {rm.x, rm.y, rm.z, rm.w};
            for (int r = 0; r < 4; ++r) {
                o0[e0 + r] = r0a[r];
                o1[e0 + r] = r1a[r];
                om[e0 + r] = rma[r];
            }
        }
    } else {
        for (int r = 0; r < 4; ++r) {
            int e = tile0 + t * 4 + r;
            if (e < E) {
                int a = relab[esrc[e]];
                int b = relab[edst[e]];
                bool v = (a >= 0) && (b >= 0);
                o0[e] = v ? (float)a : -1.f;
                o1[e] = v ? (float)b : -1.f;
                om[e] = v ? 1.f : 0.f;
            }
        }
    }
}

// ---------------------------------------------------------------------------
// Host orchestration (graph-capture safe: only kernel launches on `stream`).
// ---------------------------------------------------------------------------
extern "C" void kernel_launch(void* const* d_in, const int* in_sizes, int n_in,
                              void* d_out, int out_size, void* d_ws, size_t ws_size,
                              hipStream_t stream) {
    const float* x    = (const float*)d_in[0];
    const int*   edge = (const int*)d_in[1];
    const float* w    = (const float*)d_in[2];
    float* out = (float*)d_out;

    int  n = in_sizes[0] / 3;            // 1,000,000 nodes
    int  E = in_sizes[1] / 2;            // 16,000,000 edges
    int  k = (n + 1) / 2;                // ceil(0.5 * n)
    int  nb = (n + SORT_TILE - 1) / SORT_TILE;   // ~489 radix blocks

    char* ws = (char*)d_ws;
    unsigned* keysA   = (unsigned*)(ws);
    int*      valsA   = (int*)(ws + (size_t)n * 4);
    unsigned* keysB   = (unsigned*)(ws + (size_t)n * 8);
    int*      valsB   = (int*)(ws + (size_t)n * 12);
    unsigned* hist    = (unsigned*)(ws + (size_t)n * 16);
    int*      new_idx = (int*)(ws + (size_t)n * 16 + (size_t)nb * 256 * 4);

    int gridN = (n + THREADS - 1) / THREADS;
    score_kernel<<<gridN, THREADS, 0, stream>>>(x, w, keysA, valsA, n);

    unsigned* kin = keysA; int* vin = valsA;
    unsigned* kout = keysB; int* vout = valsB;
    for (int p = 0; p < 4; ++p) {
        int shift = p * 8;
        hist_kernel<<<nb, 256, 0, stream>>>(kin, hist, n, nb, shift);
        scan_kernel<<<1, 256, 0, stream>>>(hist, nb);
        scatter_kernel<<<nb, 256, 0, stream>>>(kin, vin, kout, vout, hist, n, nb, shift);
        unsigned* tk = kin; kin = kout; kout = tk;
        int* tv = vin; vin = vout; vout = tv;
    }
    // after 4 passes, sorted (key,val) is back in keysA/valsA (== kin/vin)
    gather_kernel<<<gridN, THREADS, 0, stream>>>(x, w, vin, out, new_idx, n, k);

    float* o0 = out + (size_t)3 * (size_t)k;
    float* o1 = o0 + (size_t)E;
    float* om = o0 + 2 * (size_t)E;
    int gridE = (E + EDGE_TILE - 1) / EDGE_TILE;
    edge_kernel<<<gridE, 256, 0, stream>>>(edge, edge + E, new_idx, o0, o1, om, E);
}